// HarmonicAngle_5454608466126
// MI455X (gfx1250) — compile-verified
//
#include <hip/hip_runtime.h>
#include <hip/hip_bf16.h>

// Harmonic angle energy: E = sum( 0.5 * k * (acos(clip(cos_theta)) - theta0)^2 )
// Memory-bound gather+reduction on MI455X:
//  - coords (12 MB) kept L2-resident (192 MB L2): gathers use default RT hints.
//  - angles/theta0/k (80 MB, read-once) streamed with non-temporal (TH=NT) b128
//    loads so they don't evict coords from near caches.
//  - global_prefetch_b8 keeps the streaming index chunks ahead of the
//    gather-latency chain.
//  - wave32 shuffle reduction + deterministic two-pass tree (graph replays must
//    be bitwise stable; float atomics would not be).

#define THREADS   256
#define GRAIN     4
#define NBLOCKS1  2048   // number of partial sums (8 KB in d_ws)

typedef __attribute__((ext_vector_type(4))) int   v4i;
typedef __attribute__((ext_vector_type(4))) float v4f;

__device__ __forceinline__ float angle_energy(const float* __restrict__ coords,
                                              int ia, int ja, int ka,
                                              float t0, float kf) {
    const float* ci = coords + 3 * ia;
    const float* cj = coords + 3 * ja;
    const float* ck = coords + 3 * ka;
    float cjx = cj[0], cjy = cj[1], cjz = cj[2];
    float v1x = ci[0] - cjx, v1y = ci[1] - cjy, v1z = ci[2] - cjz;
    float v2x = ck[0] - cjx, v2y = ck[1] - cjy, v2z = ck[2] - cjz;
    float dot = fmaf(v1x, v2x, fmaf(v1y, v2y, v1z * v2z));
    float m1  = fmaf(v1x, v1x, fmaf(v1y, v1y, v1z * v1z));
    float m2  = fmaf(v2x, v2x, fmaf(v2y, v2y, v2z * v2z));
    float c   = dot / sqrtf(m1 * m2);
    c = fminf(1.0f, fmaxf(-1.0f, c));
    float d = acosf(c) - t0;
    return 0.5f * kf * d * d;
}

__global__ __launch_bounds__(THREADS)
void angle_partial_kernel(const float* __restrict__ coords,
                          const int*   __restrict__ angles,   // [n][3] int32
                          const float* __restrict__ theta0,
                          const float* __restrict__ kvals,
                          float*       __restrict__ partial,
                          int n) {
    const int tid    = threadIdx.x;
    const int stride = gridDim.x * blockDim.x * GRAIN;   // angles per grid pass
    float acc = 0.0f;

    for (int base = (blockIdx.x * blockDim.x + tid) * GRAIN; base < n; base += stride) {
        // Speculative prefetch of next pass's chunks (gfx1250 global_prefetch_b8;
        // OOB speculative prefetches are silently dropped per ISA).
        __builtin_prefetch(angles + (size_t)(base + stride) * 3, 0, 1);
        __builtin_prefetch(theta0 + (base + stride), 0, 1);

        if (base + GRAIN <= n) {
            // 4 angles -> 12 ints: 48 bytes, 16B-aligned (base % 4 == 0) -> 3x b128 NT.
            const v4i* ap = (const v4i*)(angles + (size_t)base * 3);
            v4i a0 = __builtin_nontemporal_load(ap + 0);
            v4i a1 = __builtin_nontemporal_load(ap + 1);
            v4i a2 = __builtin_nontemporal_load(ap + 2);
            v4f t4 = __builtin_nontemporal_load((const v4f*)(theta0 + base));
            v4f k4 = __builtin_nontemporal_load((const v4f*)(kvals  + base));

            acc += angle_energy(coords, a0.x, a0.y, a0.z, t4.x, k4.x);
            acc += angle_energy(coords, a0.w, a1.x, a1.y, t4.y, k4.y);
            acc += angle_energy(coords, a1.z, a1.w, a2.x, t4.z, k4.z);
            acc += angle_energy(coords, a2.y, a2.z, a2.w, t4.w, k4.w);
        } else {
            for (int e = base; e < n; ++e) {
                int ia = angles[3 * e + 0];
                int ja = angles[3 * e + 1];
                int ka = angles[3 * e + 2];
                acc += angle_energy(coords, ia, ja, ka, theta0[e], kvals[e]);
            }
        }
    }

    // Intra-wave reduction (wave32 on CDNA5).
    #pragma unroll
    for (int off = 16; off > 0; off >>= 1)
        acc += __shfl_down(acc, off, 32);

    __shared__ float smem[THREADS / 32];
    const int lane = tid & 31;
    const int wave = tid >> 5;
    if (lane == 0) smem[wave] = acc;
    __syncthreads();

    if (wave == 0) {
        acc = (lane < (THREADS / 32)) ? smem[lane] : 0.0f;
        #pragma unroll
        for (int off = (THREADS / 64); off > 0; off >>= 1)
            acc += __shfl_down(acc, off, 32);
        if (lane == 0) partial[blockIdx.x] = acc;   // every block writes -> d_ws fully initialized
    }
}

__global__ __launch_bounds__(THREADS)
void final_reduce_kernel(const float* __restrict__ partial, int nb,
                         float* __restrict__ out) {
    const int tid = threadIdx.x;
    float acc = 0.0f;
    for (int i = tid; i < nb; i += THREADS)   // fixed order -> deterministic
        acc += partial[i];

    #pragma unroll
    for (int off = 16; off > 0; off >>= 1)
        acc += __shfl_down(acc, off, 32);

    __shared__ float smem[THREADS / 32];
    const int lane = tid & 31;
    const int wave = tid >> 5;
    if (lane == 0) smem[wave] = acc;
    __syncthreads();

    if (wave == 0) {
        acc = (lane < (THREADS / 32)) ? smem[lane] : 0.0f;
        #pragma unroll
        for (int off = (THREADS / 64); off > 0; off >>= 1)
            acc += __shfl_down(acc, off, 32);
        if (lane == 0) out[0] = acc;
    }
}

extern "C" void kernel_launch(void* const* d_in, const int* in_sizes, int n_in,
                              void* d_out, int out_size, void* d_ws, size_t ws_size,
                              hipStream_t stream) {
    const float* coords = (const float*)d_in[0];   // [N_ATOMS,3] f32
    const int*   angles = (const int*)  d_in[1];   // [N_ANGLES,3] int32 (JAX x64-off)
    const float* theta0 = (const float*)d_in[2];   // [N_ANGLES] f32
    const float* kvals  = (const float*)d_in[3];   // [N_ANGLES] f32
    const int n = in_sizes[2];                     // N_ANGLES

    float* partial = (float*)d_ws;                 // NBLOCKS1 floats (8 KB)

    angle_partial_kernel<<<NBLOCKS1, THREADS, 0, stream>>>(coords, angles, theta0,
                                                           kvals, partial, n);
    final_reduce_kernel<<<1, THREADS, 0, stream>>>(partial, NBLOCKS1, (float*)d_out);
}